// AttentionBeforeConvolution_85143431676273
// MI455X (gfx1250) — compile-verified
//
#include <hip/hip_runtime.h>

// Problem dims (fixed by the reference)
#define B_ 16
#define L_ 256
#define D_ 256
#define H_ 128

typedef __attribute__((ext_vector_type(2))) float v2f;
typedef __attribute__((ext_vector_type(8))) float v8f;

// tanh: prefer the CDNA5 hardware v_tanh_f32; fall back to exp2+rcp.
__device__ __forceinline__ float fast_tanh(float x) {
#if __has_builtin(__builtin_amdgcn_tanhf)
    return __builtin_amdgcn_tanhf(x);
#else
    float t = __builtin_amdgcn_exp2f(x * 2.885390081777927f); // 2*log2(e)
    return 1.0f - 2.0f * __builtin_amdgcn_rcpf(t + 1.0f);
#endif
}
__device__ __forceinline__ float fast_exp(float x) {
    return __builtin_amdgcn_exp2f(x * 1.4426950408889634f);   // log2(e)
}

// ---------------------------------------------------------------------------
// Kernel 1: projections Wx = x@W, Ux = x@U  (M=B*L=4096, N=H=128, K=D=256)
// One wave32 per 16(M) x 64(N) strip: 4 fp32-WMMA accumulators sharing each
// A fragment (4 independent WMMAs per K-step -> latency hiding, 4x A reuse).
// ---------------------------------------------------------------------------
__global__ __launch_bounds__(32)
void proj_kernel(const float* __restrict__ x,
                 const float* __restrict__ W,
                 const float* __restrict__ U,
                 float* __restrict__ Wx,
                 float* __restrict__ Ux) {
    const int lane = threadIdx.x;            // 0..31
    const int m0   = blockIdx.x * 16;        // row tile over B*L
    const int n0   = blockIdx.y * 64;        // 64-wide col strip over H
    const float* Bm = blockIdx.z ? U : W;
    float*       Om = blockIdx.z ? Ux : Wx;

    const int mrow = m0 + (lane & 15);       // A: lane -> M = lane%16
    const int ncol = n0 + (lane & 15);       // B: lane -> N = lane%16 (+16*t)
    const int kb   = (lane >> 4) << 1;       // K sub-offset: 0 / 2

    v8f acc0 = {}, acc1 = {}, acc2 = {}, acc3 = {};
    #pragma unroll 4
    for (int k = 0; k < D_; k += 4) {
        v2f a;
        a.x = x[mrow * D_ + k + kb];         // A VGPR0: K = kb   (merges to b64)
        a.y = x[mrow * D_ + k + kb + 1];     // A VGPR1: K = kb+1
        const float* brow0 = Bm + (k + kb) * H_ + ncol;
        const float* brow1 = Bm + (k + kb + 1) * H_ + ncol;
        v2f b0, b1, b2, b3;
        b0.x = brow0[0];  b0.y = brow1[0];
        b1.x = brow0[16]; b1.y = brow1[16];
        b2.x = brow0[32]; b2.y = brow1[32];
        b3.x = brow0[48]; b3.y = brow1[48];
        acc0 = __builtin_amdgcn_wmma_f32_16x16x4_f32(false, a, false, b0, (short)0, acc0, false, false);
        acc1 = __builtin_amdgcn_wmma_f32_16x16x4_f32(false, a, false, b1, (short)0, acc1, false, false);
        acc2 = __builtin_amdgcn_wmma_f32_16x16x4_f32(false, a, false, b2, (short)0, acc2, false, false);
        acc3 = __builtin_amdgcn_wmma_f32_16x16x4_f32(false, a, false, b3, (short)0, acc3, false, false);
    }

    // C/D layout: VGPR r -> M = r + 8*(lane>=16), N = lane%16
    const int rbase = m0 + ((lane >> 4) << 3);
    #pragma unroll
    for (int r = 0; r < 8; ++r) {
        float* orow = Om + (rbase + r) * H_ + ncol;
        orow[0]  = acc0[r];
        orow[16] = acc1[r];
        orow[32] = acc2[r];
        orow[48] = acc3[r];
    }
}

// ---------------------------------------------------------------------------
// Kernel 2: per (b,i): e_j = sum_h V[h]*tanh(Wx[i,h]+Ux[j,h]); softmax over j;
// zero the diagonal (j==i excluded from context); copy x into out[:, :D].
// float4-vectorized inner loop: 1 global b128 + 2 ds b128 per 4 tanh evals.
// ---------------------------------------------------------------------------
__global__ __launch_bounds__(256)
void score_kernel(const float* __restrict__ x,
                  const float* __restrict__ Wx,
                  const float* __restrict__ Ux,
                  const float* __restrict__ V,
                  float* __restrict__ alpha,
                  float* __restrict__ out) {
    const int b   = blockIdx.x >> 8;   // / L_
    const int i   = blockIdx.x & 255;  // % L_
    const int tid = threadIdx.x;       // 0..255 (== j, == d)

    __shared__ float4 sWxi[H_ / 4];
    __shared__ float4 sV[H_ / 4];
    __shared__ float  red[256];

    if (tid < H_ / 4) {
        sWxi[tid] = reinterpret_cast<const float4*>(Wx + (size_t)(b * L_ + i) * H_)[tid];
        sV[tid]   = reinterpret_cast<const float4*>(V)[tid];
    }
    // out[b,i,0:D] = x[b,i,:]
    out[(size_t)(b * L_ + i) * (2 * D_) + tid] = x[(size_t)(b * L_ + i) * D_ + tid];
    __syncthreads();

    const float4* uxj = reinterpret_cast<const float4*>(Ux + (size_t)(b * L_ + tid) * H_);
    float e = 0.f;
    #pragma unroll 4
    for (int h = 0; h < H_ / 4; ++h) {
        const float4 u  = uxj[h];
        const float4 wv = sWxi[h];
        const float4 vv = sV[h];
        e = fmaf(vv.x, fast_tanh(wv.x + u.x), e);
        e = fmaf(vv.y, fast_tanh(wv.y + u.y), e);
        e = fmaf(vv.z, fast_tanh(wv.z + u.z), e);
        e = fmaf(vv.w, fast_tanh(wv.w + u.w), e);
    }

    // block-wide max
    red[tid] = e;
    __syncthreads();
    for (int s = 128; s > 0; s >>= 1) {
        if (tid < s) red[tid] = fmaxf(red[tid], red[tid + s]);
        __syncthreads();
    }
    const float m = red[0];
    __syncthreads();

    // block-wide sum of exp
    const float p = fast_exp(e - m);
    red[tid] = p;
    __syncthreads();
    for (int s = 128; s > 0; s >>= 1) {
        if (tid < s) red[tid] += red[tid + s];
        __syncthreads();
    }
    float a = p / red[0];
    if (tid == i) a = 0.f;  // exclude j==i from the context sum
    alpha[(size_t)(b * L_ + i) * L_ + tid] = a;
}

// ---------------------------------------------------------------------------
// Kernel 3: context c[b] = alpha'[b] @ x[b]  (16 batched 256x256x256 GEMMs),
// written into out[:, :, D:2D]. Same 16x64-strip fp32-WMMA tiling.
// ---------------------------------------------------------------------------
__global__ __launch_bounds__(32)
void ctx_kernel(const float* __restrict__ x,
                const float* __restrict__ alpha,
                float* __restrict__ out) {
    const int lane = threadIdx.x;
    const int m0   = blockIdx.x * 16;  // i tile over L
    const int n0   = blockIdx.y * 64;  // 64-wide d strip over D
    const int b    = blockIdx.z;

    const float* A  = alpha + (size_t)b * L_ * L_;
    const float* Bx = x     + (size_t)b * L_ * D_;

    const int mrow = m0 + (lane & 15);
    const int ncol = n0 + (lane & 15);
    const int kb   = (lane >> 4) << 1;

    v8f acc0 = {}, acc1 = {}, acc2 = {}, acc3 = {};
    #pragma unroll 4
    for (int k = 0; k < L_; k += 4) {
        v2f a;
        a.x = A[mrow * L_ + k + kb];
        a.y = A[mrow * L_ + k + kb + 1];
        const float* brow0 = Bx + (k + kb) * D_ + ncol;
        const float* brow1 = Bx + (k + kb + 1) * D_ + ncol;
        v2f b0, b1, b2, b3;
        b0.x = brow0[0];  b0.y = brow1[0];
        b1.x = brow0[16]; b1.y = brow1[16];
        b2.x = brow0[32]; b2.y = brow1[32];
        b3.x = brow0[48]; b3.y = brow1[48];
        acc0 = __builtin_amdgcn_wmma_f32_16x16x4_f32(false, a, false, b0, (short)0, acc0, false, false);
        acc1 = __builtin_amdgcn_wmma_f32_16x16x4_f32(false, a, false, b1, (short)0, acc1, false, false);
        acc2 = __builtin_amdgcn_wmma_f32_16x16x4_f32(false, a, false, b2, (short)0, acc2, false, false);
        acc3 = __builtin_amdgcn_wmma_f32_16x16x4_f32(false, a, false, b3, (short)0, acc3, false, false);
    }

    const int rbase = m0 + ((lane >> 4) << 3);
    #pragma unroll
    for (int r = 0; r < 8; ++r) {
        float* orow = out + (size_t)(b * L_ + rbase + r) * (2 * D_) + D_ + ncol;
        orow[0]  = acc0[r];
        orow[16] = acc1[r];
        orow[32] = acc2[r];
        orow[48] = acc3[r];
    }
}

// ---------------------------------------------------------------------------
extern "C" void kernel_launch(void* const* d_in, const int* in_sizes, int n_in,
                              void* d_out, int out_size, void* d_ws, size_t ws_size,
                              hipStream_t stream) {
    (void)in_sizes; (void)n_in; (void)out_size; (void)ws_size;

    const float* x = (const float*)d_in[0];
    const float* W = (const float*)d_in[1];
    const float* U = (const float*)d_in[2];
    const float* V = (const float*)d_in[3];
    float* out = (float*)d_out;

    float* ws    = (float*)d_ws;
    float* Wx    = ws;                                  // B*L*H floats (2 MB)
    float* Ux    = ws + (size_t)B_ * L_ * H_;           // B*L*H floats (2 MB)
    float* alpha = ws + (size_t)2 * B_ * L_ * H_;       // B*L*L floats (4 MB)

    dim3 g1(B_ * L_ / 16, H_ / 64, 2);                  // 256 x 2 x 2 waves
    proj_kernel<<<g1, 32, 0, stream>>>(x, W, U, Wx, Ux);

    score_kernel<<<B_ * L_, 256, 0, stream>>>(x, Wx, Ux, V, alpha, out);

    dim3 g3(L_ / 16, D_ / 64, B_);                      // 16 x 4 x 16 waves
    ctx_kernel<<<g3, 32, 0, stream>>>(x, alpha, out);
}